// ScatterNd_13889924235926
// MI455X (gfx1250) — compile-verified
//
#include <hip/hip_runtime.h>
#include <hip/hip_bf16.h>

// ScatterNd set: voxel (400,400,8,64) f32 -> out, then out[ix,iy,iz,:] = pixels[n,:].
// Pure data movement (~726 MB => ~31 us floor @ 23.3 TB/s). Strategy:
//   - b128 vector memory ops, fully coalesced
//   - non-temporal cache hints (TH=NT) for all zero-reuse streams: the 655 MB
//     copy working set exceeds the 192 MB L2, so don't thrash it
//   - wave32-aware sizing: 256-thread blocks = 8 waves, ~65k waves in flight

typedef float v4f __attribute__((ext_vector_type(4)));

#define SN_Y 400
#define SN_Z 8
#define SN_C 64           // 64 f32 per voxel cell
#define SN_C4 (SN_C / 4)  // 16 x float4

// ---------------------------------------------------------------------------
// Kernel 1: bulk copy voxel -> out. NT loads + NT stores (streaming, no reuse).
// 4-way unrolled grid-stride for memory-level parallelism.
// ---------------------------------------------------------------------------
__global__ void __launch_bounds__(256)
sn_copy_b128(const v4f* __restrict__ src, v4f* __restrict__ dst, long n4) {
    const long stride = (long)gridDim.x * blockDim.x;
    long i = (long)blockIdx.x * blockDim.x + threadIdx.x;

    for (; i + 3 * stride < n4; i += 4 * stride) {
        v4f a = __builtin_nontemporal_load(src + i);
        v4f b = __builtin_nontemporal_load(src + i + stride);
        v4f c = __builtin_nontemporal_load(src + i + 2 * stride);
        v4f d = __builtin_nontemporal_load(src + i + 3 * stride);
        __builtin_nontemporal_store(a, dst + i);
        __builtin_nontemporal_store(b, dst + i + stride);
        __builtin_nontemporal_store(c, dst + i + 2 * stride);
        __builtin_nontemporal_store(d, dst + i + 3 * stride);
    }
    for (; i < n4; i += stride) {
        v4f a = __builtin_nontemporal_load(src + i);
        __builtin_nontemporal_store(a, dst + i);
    }
}

// ---------------------------------------------------------------------------
// Kernel 2: scatter. One thread per (pixel, float4-chunk): 16 consecutive lanes
// own one pixel row (64 f32 = 256 B) => fully-coalesced contiguous 256 B read
// from pixels (NT, no reuse) and contiguous aligned 256 B write into out.
// Index triples are shared by 16 lanes -> regular (RT) loads to exploit cache.
// ---------------------------------------------------------------------------
__global__ void __launch_bounds__(256)
sn_scatter_b128(const int* __restrict__ idx,
                const v4f* __restrict__ pix4,
                v4f* __restrict__ out4,
                int n_pixels) {
    const int gid   = blockIdx.x * blockDim.x + threadIdx.x;
    const int total = n_pixels * SN_C4;
    if (gid >= total) return;

    const int p = gid >> 4;           // pixel id
    const int c = gid & (SN_C4 - 1);  // float4 chunk within the row

    const int ix = idx[3 * p + 0];
    const int iy = idx[3 * p + 1];
    const int iz = idx[3 * p + 2];

    // cell base in float4 units: ((ix*Y + iy)*Z + iz) * (C/4); always 256B-aligned
    const long cell  = ((long)ix * SN_Y + iy) * SN_Z + iz;
    const long base4 = cell * SN_C4;

    v4f v = __builtin_nontemporal_load(pix4 + (long)p * SN_C4 + c);
    __builtin_nontemporal_store(v, out4 + base4 + c);
}

// ---------------------------------------------------------------------------
// d_in[0]=voxel f32, d_in[1]=scatter_indices i32 (N,3), d_in[2]=pixels f32
// ---------------------------------------------------------------------------
extern "C" void kernel_launch(void* const* d_in, const int* in_sizes, int n_in,
                              void* d_out, int out_size, void* d_ws, size_t ws_size,
                              hipStream_t stream) {
    const float* voxel = (const float*)d_in[0];
    const int*   sidx  = (const int*)d_in[1];
    const float* pix   = (const float*)d_in[2];
    float*       out   = (float*)d_out;

    const long n4       = (long)out_size / 4;  // 20,480,000 float4
    const int  n_pixels = in_sizes[1] / 3;     // 134,400

    // 1) bulk copy: rewrites all of d_out every call -> deterministic replays
    {
        const int threads = 256;
        const int blocks  = 8192;  // ~65k wave32 waves: saturates HBM
        sn_copy_b128<<<blocks, threads, 0, stream>>>(
            (const v4f*)voxel, (v4f*)out, n4);
    }

    // 2) scatter overwrite (same stream -> ordered after the copy)
    {
        const int threads = 256;
        const int total   = n_pixels * SN_C4;  // 2,150,400 threads
        const int blocks  = (total + threads - 1) / threads;
        sn_scatter_b128<<<blocks, threads, 0, stream>>>(
            sidx, (const v4f*)pix, (v4f*)out, n_pixels);
    }
}